// GwACGraph_28123445854580
// MI455X (gfx1250) — compile-verified
//
#include <hip/hip_runtime.h>
#include <hip/hip_bf16.h>
#include <math.h>

typedef __attribute__((ext_vector_type(2))) float v2f;
typedef __attribute__((ext_vector_type(8))) float v8f;

#define NN      2000
#define IN_F    32
#define HID     64
#define OUT_F   16
#define MSGW    32
#define START0  1000
#define ROW_PAD 68   // LDS row pitch in floats: (4*m + c) % 64 distinct per row -> no bank conflicts

__global__ __launch_bounds__(32)
void gwac_fused_kernel(const float* __restrict__ x,
                       const float* __restrict__ We,  const float* __restrict__ be,
                       const float* __restrict__ Wns, const float* __restrict__ bns,
                       const float* __restrict__ Wd,  const float* __restrict__ bd,
                       float* __restrict__ out)
{
    __shared__ float encS[16 * ROW_PAD];
    __shared__ float hS[16 * ROW_PAD];
    __shared__ float cvecS[HID];
    __shared__ float logitS[16 * 17];

    const int lane = threadIdx.x;            // 0..31 (wave32)
    const int tile = blockIdx.x;             // 0..124, rows [tile*16, tile*16+16)
    const int hi   = (lane >= 16) ? 1 : 0;   // which K-half of a K=4 chunk this lane holds
    const int l16  = lane & 15;
    const int kOff = hi * 2;
    const int mRow = l16;                    // A-fragment row owned by this lane

    // cvec[n] = bns[n] + sum_j Wns[n][64+j]   (message part contracted with ones(32))
    #pragma unroll
    for (int rep = 0; rep < 2; ++rep) {
        int n = lane + rep * 32;
        float s = bns[n];
        const float* wr = Wns + n * (HID + MSGW) + HID;
        #pragma unroll
        for (int j = 0; j < MSGW; ++j) s += wr[j];
        cvecS[n] = s;
    }

    // ---------------- GEMM1: enc = Xtile(16x32) @ We^T + be ----------------
    v2f a1[8];
    {
        const float* xr = x + (tile * 16 + mRow) * IN_F + kOff;
        #pragma unroll
        for (int kk = 0; kk < 8; ++kk) {
            a1[kk].x = xr[4 * kk + 0];
            a1[kk].y = xr[4 * kk + 1];
        }
    }
    #pragma unroll
    for (int nt = 0; nt < 4; ++nt) {
        v8f acc = {};
        const float* wrow = We + (nt * 16 + l16) * IN_F + kOff;   // column n = nt*16+l16
        #pragma unroll
        for (int kk = 0; kk < 8; ++kk) {
            v2f b; b.x = wrow[4 * kk + 0]; b.y = wrow[4 * kk + 1];
            acc = __builtin_amdgcn_wmma_f32_16x16x4_f32(false, a1[kk], false, b,
                                                        (short)0, acc, false, false);
        }
        float bias = be[nt * 16 + l16];
        #pragma unroll
        for (int r = 0; r < 8; ++r) {
            int M = r + 8 * hi;                                   // C/D layout
            encS[M * ROW_PAD + nt * 16 + l16] = acc[r] + bias;
        }
    }
    __syncthreads();

    // ---------------- GEMM2: h = relu(enc(16x64) @ WnsL^T + cvec), masked ----------------
    v2f a2[16];
    #pragma unroll
    for (int kk = 0; kk < 16; ++kk) {
        a2[kk].x = encS[mRow * ROW_PAD + 4 * kk + kOff + 0];
        a2[kk].y = encS[mRow * ROW_PAD + 4 * kk + kOff + 1];
    }
    #pragma unroll
    for (int nt = 0; nt < 4; ++nt) {
        v8f acc = {};
        const float* wrow = Wns + (nt * 16 + l16) * (HID + MSGW) + kOff;
        #pragma unroll
        for (int kk = 0; kk < 16; ++kk) {
            v2f b; b.x = wrow[4 * kk + 0]; b.y = wrow[4 * kk + 1];
            acc = __builtin_amdgcn_wmma_f32_16x16x4_f32(false, a2[kk], false, b,
                                                        (short)0, acc, false, false);
        }
        float cv = cvecS[nt * 16 + l16];
        #pragma unroll
        for (int r = 0; r < 8; ++r) {
            int M = r + 8 * hi;
            int grow = tile * 16 + M;
            float t = fmaxf(acc[r] + cv, 0.0f);
            hS[M * ROW_PAD + nt * 16 + l16] = (grow >= START0) ? t : 0.0f; // non-starts keep zero state
        }
    }
    __syncthreads();

    // ---------------- GEMM3: logits = h(16x64) @ Wd^T + bd ----------------
    v8f acc3 = {};
    {
        const float* wrow = Wd + l16 * HID + kOff;                // column n = l16 (OUT_F = 16)
        #pragma unroll
        for (int kk = 0; kk < 16; ++kk) {
            v2f a, b;
            a.x = hS[mRow * ROW_PAD + 4 * kk + kOff + 0];
            a.y = hS[mRow * ROW_PAD + 4 * kk + kOff + 1];
            b.x = wrow[4 * kk + 0];
            b.y = wrow[4 * kk + 1];
            acc3 = __builtin_amdgcn_wmma_f32_16x16x4_f32(false, a, false, b,
                                                         (short)0, acc3, false, false);
        }
    }
    {
        float bdv = bd[l16];
        #pragma unroll
        for (int r = 0; r < 8; ++r) {
            int M = r + 8 * hi;
            logitS[M * 17 + l16] = acc3[r] + bdv;
        }
    }
    __syncthreads();

    // ---------------- row-wise log_softmax over 16 classes ----------------
    if (lane < 16) {
        int M = lane;
        int grow = tile * 16 + M;
        float v[16];
        float mx = -INFINITY;
        #pragma unroll
        for (int n = 0; n < 16; ++n) { v[n] = logitS[M * 17 + n]; mx = fmaxf(mx, v[n]); }
        float s = 0.0f;
        #pragma unroll
        for (int n = 0; n < 16; ++n) s += expf(v[n] - mx);
        float lse = mx + logf(s);
        float* orow = out + grow * OUT_F;
        #pragma unroll
        for (int n = 0; n < 16; ++n) orow[n] = v[n] - lse;
    }
}

extern "C" void kernel_launch(void* const* d_in, const int* in_sizes, int n_in,
                              void* d_out, int out_size, void* d_ws, size_t ws_size,
                              hipStream_t stream) {
    (void)in_sizes; (void)n_in; (void)out_size; (void)d_ws; (void)ws_size;
    // input order: x, nbr, deg, We, be, Wns, bns, Wnm, bnm, Wd, bd
    const float* x   = (const float*)d_in[0];
    const float* We  = (const float*)d_in[3];
    const float* be  = (const float*)d_in[4];
    const float* Wns = (const float*)d_in[5];
    const float* bns = (const float*)d_in[6];
    const float* Wd  = (const float*)d_in[9];
    const float* bd  = (const float*)d_in[10];
    float* out = (float*)d_out;

    // 125 tiles of 16 rows cover all 2000 nodes; one wave32 per tile.
    gwac_fused_kernel<<<dim3(NN / 16), dim3(32), 0, stream>>>(x, We, be, Wns, bns, Wd, bd, out);
}